// DenseCaptioningLoss_79130477462087
// MI455X (gfx1250) — compile-verified
//
#include <hip/hip_runtime.h>
#include <math.h>

// Problem dimensions (from reference)
#define BS   8
#define MAXC 16
#define CLEN 32
#define CV   12000
#define PLEN 128
#define PV   512
#define SEM  300

typedef __attribute__((ext_vector_type(2))) float v2f;
typedef __attribute__((ext_vector_type(4))) float v4f;
typedef __attribute__((ext_vector_type(8))) float v8f;

// ---------------------------------------------------------------------------
// Exact f32 sum of 256 per-thread partials using V_WMMA_F32_16X16X4_F32.
// A = ones(16x4), B = 4x16 chunk of data (64 values), D = A*B + C gives the
// column sums of B replicated across all 16 rows of D, accumulated in f32.
// 4 chained WMMAs consume all 256 values; then a 32-lane collapse of D's
// first VGPR (rows M=0 and M=8, each holding identical column sums) gives
// exactly 2x the total.  All math is f32 -> bit-compatible with VALU adds.
// Returns the total to ALL threads of the block.
// ---------------------------------------------------------------------------
__device__ __forceinline__ float block_sum_256(float v, float* sm) {
    const int tid = threadIdx.x;
    sm[tid] = v;
    __syncthreads();
    if (tid < 32) {                       // wave 0 only: EXEC all-ones in-wave
        v2f a; a.x = 1.0f; a.y = 1.0f;    // A = ones(16x4)
        v8f acc = {};
        #pragma unroll
        for (int j = 0; j < 4; ++j) {
            v2f b;
            b.x = sm[j * 64 + tid];
            b.y = sm[j * 64 + 32 + tid];
            acc = __builtin_amdgcn_wmma_f32_16x16x4_f32(
                /*neg_a=*/false, a, /*neg_b=*/false, b,
                /*c_mod=*/(short)0, acc, /*reuse_a=*/false, /*reuse_b=*/false);
        }
        float r = acc[0];                 // lanes 0-15: D[0][n]; 16-31: D[8][n]
        #pragma unroll
        for (int off = 16; off > 0; off >>= 1)
            r += __shfl_xor(r, off, 32);  // = 2 * total (exact)
        if (tid == 0) sm[0] = 0.5f * r;
    }
    __syncthreads();
    float total = sm[0];
    __syncthreads();
    return total;
}

__device__ __forceinline__ float block_max_256(float v, float* sm) {
    const int tid = threadIdx.x;
    sm[tid] = v;
    __syncthreads();
    #pragma unroll
    for (int off = 128; off > 0; off >>= 1) {
        if (tid < off) sm[tid] = fmaxf(sm[tid], sm[tid + off]);
        __syncthreads();
    }
    float m = sm[0];
    __syncthreads();
    return m;
}

// ---------------------------------------------------------------------------
// Kernel 1: one 256-thread block per caption row (4096 rows x 12000 logits).
// Single-pass online softmax: the 196.6 MB tensor is read once, as coalesced
// non-temporal b128 loads (stream-once data; keep the 192 MB L2 clean).
// Online update uses exactly ONE v_exp_f32 per element:
//   t = exp(-|x-m|);  x<=m: s += t;  x>m: s = s*t + 1, m = x.
// ---------------------------------------------------------------------------
__global__ __launch_bounds__(256)
void cap_row_kernel(const float* __restrict__ logits,
                    const int*   __restrict__ gt,
                    const int*   __restrict__ cap_lens,
                    const int*   __restrict__ caps_count,
                    float* __restrict__ nll_out,
                    float* __restrict__ mask_out) {
    __shared__ float sm[256];
    const int row = blockIdx.x;                 // b*MAXC*CLEN + c*CLEN + t
    const int b = row / (MAXC * CLEN);
    const int c = (row / CLEN) % MAXC;
    const int t = row % CLEN;
    const int tid = threadIdx.x;
    const float* p  = logits + (size_t)row * CV;
    const v4f*   p4 = (const v4f*)p;            // CV % 4 == 0

    float m = -INFINITY, s = 0.0f;
    for (int i = tid; i < CV / 4; i += 256) {
        v4f x = __builtin_nontemporal_load(&p4[i]);
        #pragma unroll
        for (int k = 0; k < 4; ++k) {
            float xv = x[k];
            float d  = xv - m;                  // first iter: -inf -> t = 0
            float e  = __expf(-fabsf(d));       // single v_exp_f32
            bool  up = d > 0.0f;
            s = up ? __builtin_fmaf(s, e, 1.0f) : (s + e);
            m = fmaxf(m, xv);
        }
    }
    // combine across block: rescale to the block max, WMMA-sum the partials
    float M  = block_max_256(m, sm);
    float sp = s * __expf(m - M);
    float S  = block_sum_256(sp, sm);

    if (tid == 0) {
        float lse = M + logf(S);
        int   g   = gt[row];
        float nll = lse - p[g];
        bool valid = (c < caps_count[b]) && (t < cap_lens[b * MAXC + c]);
        nll_out[row]  = valid ? nll : 0.0f;
        mask_out[row] = valid ? 1.0f : 0.0f;
    }
}

// ---------------------------------------------------------------------------
// Kernel 2: one block per program row (1024 rows x 512 logits), float2/lane.
// ---------------------------------------------------------------------------
__global__ __launch_bounds__(256)
void prog_row_kernel(const float* __restrict__ logits,
                     const int*   __restrict__ gt,
                     const int*   __restrict__ prog_len,
                     float* __restrict__ nll_out,
                     float* __restrict__ mask_out) {
    __shared__ float sm[256];
    const int row = blockIdx.x;                 // b*PLEN + t
    const int b = row / PLEN;
    const int t = row % PLEN;
    const int tid = threadIdx.x;
    const float* p  = logits + (size_t)row * PV;
    const v2f*   p2 = (const v2f*)p;
    v2f x = p2[tid];                            // 256 threads x float2 = 512
    float m = fmaxf(x.x, x.y);
    float M = block_max_256(m, sm);
    float sp = __expf(x.x - M) + __expf(x.y - M);
    float S  = block_sum_256(sp, sm);
    if (tid == 0) {
        float lse = M + logf(S);
        int   g   = gt[row];
        bool valid = (t < prog_len[b]);
        nll_out[row]  = valid ? (lse - p[g]) : 0.0f;
        mask_out[row] = valid ? 1.0f : 0.0f;
    }
}

// ---------------------------------------------------------------------------
// Kernel 3: one block per caption slot (128 slots): BCE over SEM=300 + tIoU.
// ---------------------------------------------------------------------------
__global__ __launch_bounds__(256)
void sem_iou_kernel(const float* __restrict__ pred_sem,
                    const float* __restrict__ gt_sem,
                    const int*   __restrict__ caps_count,
                    const float* __restrict__ pred_iv,
                    const float* __restrict__ gt_iv,
                    float* __restrict__ bce_out,
                    float* __restrict__ cm_out,
                    float* __restrict__ iou_out) {
    __shared__ float sm[256];
    const int row = blockIdx.x;                 // b*MAXC + c
    const int b = row / MAXC;
    const int c = row % MAXC;
    const int tid = threadIdx.x;
    const float* p = pred_sem + (size_t)row * SEM;
    const float* y = gt_sem   + (size_t)row * SEM;
    float acc = 0.0f;
    for (int i = tid; i < SEM; i += 256) {
        float pi = p[i], yi = y[i];
        acc -= yi * logf(pi) + (1.0f - yi) * logf(1.0f - pi);
    }
    float S = block_sum_256(acc, sm);
    if (tid == 0) {
        float valid = (c < caps_count[b]) ? 1.0f : 0.0f;
        bce_out[row] = S * valid;
        cm_out[row]  = valid;
        float ps = pred_iv[row * 2], pe = pred_iv[row * 2 + 1];
        float gs = gt_iv[row * 2],   ge = gt_iv[row * 2 + 1];
        float inter = fmaxf(fminf(pe, ge) - fmaxf(ps, gs), 0.0f);
        float uni   = fmaxf(pe, ge) - fminf(ps, gs);
        iou_out[row] = (inter / uni) * valid;
    }
}

// ---------------------------------------------------------------------------
// Kernel 4: single block combines everything into the 5 output scalars.
// ---------------------------------------------------------------------------
__global__ __launch_bounds__(256)
void finalize_kernel(const float* __restrict__ nllc, const float* __restrict__ maskc,
                     const float* __restrict__ nllp, const float* __restrict__ maskp,
                     const float* __restrict__ bce,  const float* __restrict__ cm,
                     const float* __restrict__ iou,  float* __restrict__ out) {
    __shared__ float sm[256];
    const int tid = threadIdx.x;

    float a = 0.0f, bmm = 0.0f;
    for (int i = tid; i < BS * MAXC * CLEN; i += 256) { a += nllc[i]; bmm += maskc[i]; }
    float sum_nllc  = block_sum_256(a,   sm);
    float sum_maskc = block_sum_256(bmm, sm);

    float c1 = 0.0f, c2 = 0.0f;
    for (int i = tid; i < BS * PLEN; i += 256) { c1 += nllp[i]; c2 += maskp[i]; }
    float sum_nllp  = block_sum_256(c1, sm);
    float sum_maskp = block_sum_256(c2, sm);

    float d1 = (tid < BS * MAXC) ? bce[tid] : 0.0f;
    float d2 = (tid < BS * MAXC) ? cm[tid]  : 0.0f;
    float d3 = (tid < BS * MAXC) ? iou[tid] : 0.0f;
    float sum_bce = block_sum_256(d1, sm);
    float n_caps  = block_sum_256(d2, sm);
    float sum_iou = block_sum_256(d3, sm);

    if (tid == 0) {
        float cap_loss  = sum_nllc / sum_maskc;
        float prog_loss = sum_nllp / sum_maskp;
        float sem_loss  = sum_bce / (n_caps * (float)SEM);
        float iou_loss  = 1.0f - sum_iou / n_caps;
        float loss      = cap_loss + prog_loss + sem_loss;
        out[0] = loss;
        out[1] = prog_loss;
        out[2] = cap_loss;
        out[3] = sem_loss;
        out[4] = iou_loss;
    }
}

// ---------------------------------------------------------------------------
extern "C" void kernel_launch(void* const* d_in, const int* in_sizes, int n_in,
                              void* d_out, int out_size, void* d_ws, size_t ws_size,
                              hipStream_t stream) {
    const float* pred_captions = (const float*)d_in[0];
    const int*   gt_captions   = (const int*)  d_in[1];
    const int*   gt_cap_lens   = (const int*)  d_in[2];
    const int*   gt_caps_count = (const int*)  d_in[3];
    const float* pred_sem      = (const float*)d_in[4];
    const float* gt_sem        = (const float*)d_in[5];
    const float* pred_program  = (const float*)d_in[6];
    const int*   gt_program    = (const int*)  d_in[7];
    const int*   gt_prog_len   = (const int*)  d_in[8];
    const float* pred_iv       = (const float*)d_in[9];
    const float* gt_iv         = (const float*)d_in[10];
    float* out = (float*)d_out;

    // workspace layout (floats)
    float* ws    = (float*)d_ws;
    float* nllc  = ws;                         // 4096
    float* maskc = nllc  + BS * MAXC * CLEN;   // 4096
    float* nllp  = maskc + BS * MAXC * CLEN;   // 1024
    float* maskp = nllp  + BS * PLEN;          // 1024
    float* bce   = maskp + BS * PLEN;          // 128
    float* cm    = bce   + BS * MAXC;          // 128
    float* iou   = cm    + BS * MAXC;          // 128

    cap_row_kernel<<<BS * MAXC * CLEN, 256, 0, stream>>>(
        pred_captions, gt_captions, gt_cap_lens, gt_caps_count, nllc, maskc);
    prog_row_kernel<<<BS * PLEN, 256, 0, stream>>>(
        pred_program, gt_program, gt_prog_len, nllp, maskp);
    sem_iou_kernel<<<BS * MAXC, 256, 0, stream>>>(
        pred_sem, gt_sem, gt_caps_count, pred_iv, gt_iv, bce, cm, iou);
    finalize_kernel<<<1, 256, 0, stream>>>(
        nllc, maskc, nllp, maskp, bce, cm, iou, out);
}